// simOTA_833223655481
// MI455X (gfx1250) — compile-verified
//
#include <hip/hip_runtime.h>
#include <hip/hip_bf16.h>
#include <math.h>

typedef __attribute__((ext_vector_type(2))) float v2f;
typedef __attribute__((ext_vector_type(8))) float v8f;

#define BS   32
#define NA   8400
#define NG   64
#define NCLS 80
#define TOPKN 10
#define EPSF 1e-9f
#define BIGF 100000.0f
#define DISF 2.5f

// ---------------------------------------------------------------------------
// Phase 1: S[b,a] = -sum_c log(1-clamp(p)) via V_WMMA_F32_16X16X4_F32.
// One wave handles 16 anchors. A-matrix 16x4 f32 layout (ISA 7.12.2):
//   lanes 0-15 hold {K=0,K=1} of row M=lane; lanes 16-31 hold {K=2,K=3}.
// B = all-ones 4x16 -> D[m,n] = row sum, replicated over n.
// D layout: VGPR r, lanes 0-15 -> M=r; lanes 16-31 -> M=8+r.
// Uses __logf (v_log_f32) instead of libm log1pf: p is clamped to
// [1e-7, 1-1e-7] so 1-p is a safe, normal argument.
// ---------------------------------------------------------------------------
__global__ __launch_bounds__(256) void rowsum_wmma_kernel(
    const float* __restrict__ scores, float* __restrict__ S)
{
    const int lane  = threadIdx.x & 31;
    const int wave  = threadIdx.x >> 5;
    const int b     = blockIdx.x;
    const int abase = (blockIdx.y * 8 + wave) * 16;
    if (abase >= NA) return;                 // wave-uniform exit

    const int m    = lane & 15;
    const int koff = (lane >> 4) * 2;
    const float* row = scores + ((size_t)b * NA + abase + m) * NCLS + koff;

    v8f acc = {0.f, 0.f, 0.f, 0.f, 0.f, 0.f, 0.f, 0.f};
    v2f ones; ones.x = 1.0f; ones.y = 1.0f;

    #pragma unroll
    for (int k0 = 0; k0 < NCLS; k0 += 4) {
        float p0 = row[k0];
        float p1 = row[k0 + 1];
        p0 = fminf(fmaxf(p0, 1e-7f), 1.0f - 1e-7f);
        p1 = fminf(fmaxf(p1, 1e-7f), 1.0f - 1e-7f);
        v2f av; av.x = -__logf(1.0f - p0); av.y = -__logf(1.0f - p1);
        acc = __builtin_amdgcn_wmma_f32_16x16x4_f32(
            false, av, false, ones, (short)0, acc, false, false);
    }

    if ((lane & 15) == 0) {                  // lane 0 -> rows 0..7, lane 16 -> rows 8..15
        float* o = S + (size_t)b * NA + abase + (lane >> 4) * 8;
        #pragma unroll
        for (int r = 0; r < 8; ++r) o[r] = acc[r];
    }
}

// ---------------------------------------------------------------------------
// Phase 2: per (b,g) block. Computes cost row, writes it to d_out, tracks
// block-wide top-10 smallest costs, scatters GT bit into per-anchor bitmask.
// Fast-rate math: __fdividef (v_rcp_f32+mul) for all divisions, single
// __logf for the class term log((1-p)/p).
// ---------------------------------------------------------------------------
__global__ __launch_bounds__(256) void cost_topk_kernel(
    const float4* __restrict__ pbox, const float4* __restrict__ gbox,
    const int* __restrict__ maskgt, const float* __restrict__ scores,
    const int* __restrict__ labels, const float* __restrict__ S,
    float* __restrict__ costo, unsigned long long* __restrict__ m64)
{
    __shared__ float sv[256];
    __shared__ int   si[256];
    __shared__ int   stop[TOPKN];

    const int tid = threadIdx.x;
    const int b   = blockIdx.x / NG;
    const int g   = blockIdx.x % NG;

    const float4 gb  = gbox[(size_t)b * NG + g];
    const int    lab = labels[b * NG + g];
    const int    mg  = maskgt[b * NG + g];

    const float gx1 = gb.x, gy1 = gb.y, gx2 = gb.z, gy2 = gb.w;
    const float gcx = (gx1 + gx2) * 0.5f, gcy = (gy1 + gy2) * 0.5f;
    const float w2  = gx2 - gx1, h2 = gy2 - gy1 + EPSF;
    const float at2 = atanf(__fdividef(w2, h2));            // hoisted (g-uniform)
    const float inv_pi2_4 = 4.0f / (3.14159265358979323846f * 3.14159265358979323846f);

    float*        crow = costo + ((size_t)b * NG + g) * NA;
    const float*  srow = S + (size_t)b * NA;
    const float*  prow = scores + (size_t)b * NA * NCLS + lab;
    const float4* pbr  = pbox + (size_t)b * NA;

    float bv[TOPKN]; int bi[TOPKN];
    #pragma unroll
    for (int j = 0; j < TOPKN; ++j) { bv[j] = INFINITY; bi[j] = 0; }

    for (int a = tid; a < NA; a += 256) {
        const float4 pb = pbr[a];
        const float pcx = (pb.x + pb.z) * 0.5f, pcy = (pb.y + pb.w) * 0.5f;
        const bool inb  = (pcx > gx1) && (pcy > gy1) && (gx2 > pcx) && (gy2 > pcy);
        const bool inc  = (pcx > gcx - DISF) && (pcy > gcy - DISF) &&
                          (gcx + DISF > pcx) && (gcy + DISF > pcy);
        const bool both = inb && inc;

        // CIoU(pred, gt)
        const float w1 = pb.z - pb.x, h1 = pb.w - pb.y + EPSF;
        const float iw = fmaxf(fminf(pb.z, gx2) - fmaxf(pb.x, gx1), 0.f);
        const float ih = fmaxf(fminf(pb.w, gy2) - fmaxf(pb.y, gy1), 0.f);
        const float inter = iw * ih;
        const float uni   = w1 * h1 + w2 * h2 - inter + EPSF;
        const float iou   = __fdividef(inter, uni);
        const float cw = fmaxf(pb.z, gx2) - fminf(pb.x, gx1);
        const float ch = fmaxf(pb.w, gy2) - fminf(pb.y, gy1);
        const float c2 = cw * cw + ch * ch + EPSF;
        const float dx = gx1 + gx2 - pb.x - pb.z, dy = gy1 + gy2 - pb.y - pb.w;
        const float d2 = (dx * dx + dy * dy) * 0.25f;
        const float dat = at2 - atanf(__fdividef(w1, h1));
        const float v   = inv_pi2_4 * dat * dat;
        const float alpha = __fdividef(v, v - iou + (1.0f + EPSF));
        const float ciou  = iou - (__fdividef(d2, c2) + v * alpha);

        // class cost: S + log((1-p)/p) at gt label (== log1mp - logp)
        float p = prow[(size_t)a * NCLS];
        p = fminf(fmaxf(p, 1e-7f), 1.0f - 1e-7f);
        const float perlab = __logf(__fdividef(1.0f - p, p));

        const float cost = srow[a] + perlab + 3.0f * ciou + (both ? 0.f : BIGF);
        crow[a] = cost;

        // register-resident sorted top-10 (unrolled -> no scratch spill)
        if (cost < bv[TOPKN - 1]) {
            bv[TOPKN - 1] = cost; bi[TOPKN - 1] = a;
            #pragma unroll
            for (int j = TOPKN - 1; j > 0; --j) {
                if (bv[j] < bv[j - 1]) {
                    float tv = bv[j]; bv[j] = bv[j - 1]; bv[j - 1] = tv;
                    int   ti = bi[j]; bi[j] = bi[j - 1]; bi[j - 1] = ti;
                }
            }
        }
    }

    if (!mg) return;   // block-uniform: pt_mask &= mask_gt -> nothing to scatter

    // 10 rounds of block argmin over per-thread sorted lists
    int ptr = 0;
    for (int r = 0; r < TOPKN; ++r) {
        sv[tid] = (ptr < TOPKN) ? bv[ptr] : INFINITY;
        si[tid] = tid;
        __syncthreads();
        for (int s = 128; s > 0; s >>= 1) {
            if (tid < s) {
                if (sv[tid + s] < sv[tid]) { sv[tid] = sv[tid + s]; si[tid] = si[tid + s]; }
            }
            __syncthreads();
        }
        const int owner = si[0];
        if (tid == owner) { stop[r] = bi[ptr]; ++ptr; }
        __syncthreads();
    }

    if (tid < TOPKN) {
        atomicOr(&m64[(size_t)b * NA + stop[tid]], 1ull << g);
    }
}

// ---------------------------------------------------------------------------
// Phase 3: per (b,a) resolve overlap via 64-bit GT bitmask; write final 0/1.
// ---------------------------------------------------------------------------
__global__ __launch_bounds__(256) void finalize_kernel(
    const unsigned long long* __restrict__ m64,
    const float* __restrict__ costo,
    float* __restrict__ finalo)
{
    const int idx = blockIdx.x * 256 + threadIdx.x;
    if (idx >= BS * NA) return;
    const int b = idx / NA;
    const int a = idx - b * NA;

    const unsigned long long m = m64[idx];
    const int cnt = __popcll(m);

    int pick = 0;
    if (cnt > 0) {
        float best = INFINITY;
        unsigned long long mm = m;
        const float* cb = costo + (size_t)b * NG * NA + a;
        while (mm) {
            const int g = __builtin_ctzll(mm);
            mm &= (mm - 1ull);
            const float c = cb[(size_t)g * NA];
            if (c < best) { best = c; pick = g; }
        }
    }
    const bool ov = (cnt > 1);

    float* fb = finalo + (size_t)b * NG * NA + a;
    #pragma unroll 8
    for (int g = 0; g < NG; ++g) {
        const bool f = ov ? (g == pick) : (((m >> g) & 1ull) != 0ull);
        fb[(size_t)g * NA] = f ? 1.0f : 0.0f;
    }
}

// ---------------------------------------------------------------------------
extern "C" void kernel_launch(void* const* d_in, const int* in_sizes, int n_in,
                              void* d_out, int out_size, void* d_ws, size_t ws_size,
                              hipStream_t stream)
{
    const float* pred_boxes = (const float*)d_in[0];   // (bs, A, 4) f32
    const float* gt_boxes   = (const float*)d_in[1];   // (bs, G, 4) f32
    const int*   mask_gt    = (const int*)d_in[2];     // (bs, G)    int
    const float* scores     = (const float*)d_in[3];   // (bs, A, NC) f32
    const int*   gt_labels  = (const int*)d_in[4];     // (bs, G)    int32

    float* out_final = (float*)d_out;                       // (bs, G, A) as 0/1 floats
    float* out_cost  = out_final + (size_t)BS * NG * NA;    // (bs, G, A) f32

    const size_t S_BYTES = (size_t)BS * NA * sizeof(float);        // 1.07 MB
    float* S = (float*)d_ws;
    unsigned long long* m64 =
        (unsigned long long*)((char*)d_ws + S_BYTES);              // 2.15 MB

    // bitmask must start at zero every call (harness doesn't re-poison)
    hipMemsetAsync(m64, 0, (size_t)BS * NA * sizeof(unsigned long long), stream);

    // Phase 1: WMMA class-sum. 525 waves of 16 anchors per batch -> 66 blocks of 8 waves.
    dim3 g1(BS, (NA / 16 + 7) / 8);
    rowsum_wmma_kernel<<<g1, 256, 0, stream>>>(scores, S);

    // Phase 2: cost + top-10 per (b,g)
    cost_topk_kernel<<<BS * NG, 256, 0, stream>>>(
        (const float4*)pred_boxes, (const float4*)gt_boxes,
        mask_gt, scores, gt_labels, S, out_cost, m64);

    // Phase 3: overlap resolution + final mask
    finalize_kernel<<<(BS * NA + 255) / 256, 256, 0, stream>>>(m64, out_cost, out_final);
}